// BasicEuclideanDistModel_11218454577602
// MI455X (gfx1250) — compile-verified
//
#include <hip/hip_runtime.h>

typedef __attribute__((ext_vector_type(2))) float v2f;
typedef __attribute__((ext_vector_type(8))) float v8f;

#define N_POINTS 2048
#define N_EVENTS 200000
#define RSAMP 10
#define TILES (N_POINTS / 16)          // 128
#define NTILEBLK (TILES * TILES)       // 16384 tile partials
#define NEVBLK 512                     // event-kernel blocks

// ---------------------------------------------------------------------------
// Per-point self terms: s0=|z|^2, s1=2 z.v, s2=|v|^2
// ---------------------------------------------------------------------------
__global__ void stats_kernel(const float* __restrict__ z0,
                             const float* __restrict__ v0,
                             float* __restrict__ s0,
                             float* __restrict__ s1,
                             float* __restrict__ s2) {
    int i = blockIdx.x * blockDim.x + threadIdx.x;
    if (i < N_POINTS) {
        float zx = z0[2 * i], zy = z0[2 * i + 1];
        float vx = v0[2 * i], vy = v0[2 * i + 1];
        s0[i] = zx * zx + zy * zy;
        s1[i] = 2.f * (zx * vx + zy * vy);
        s2[i] = vx * vx + vy * vy;
    }
}

// ---------------------------------------------------------------------------
// Event term: sum over events of (b - ||dz + dv * t||); fixed-order partials
// ---------------------------------------------------------------------------
__global__ void event_kernel(const float* __restrict__ beta,
                             const float* __restrict__ z0,
                             const float* __restrict__ v0,
                             const int* __restrict__ uv,
                             const float* __restrict__ tt,
                             float* __restrict__ partial) {
    __shared__ float red[256];
    float b = beta[0];
    float acc = 0.f;
    for (int e = blockIdx.x * blockDim.x + threadIdx.x; e < N_EVENTS;
         e += gridDim.x * blockDim.x) {
        int u = uv[2 * e], v = uv[2 * e + 1];
        float t = tt[e];
        float dx = (z0[2 * u]     - z0[2 * v])     + (v0[2 * u]     - v0[2 * v])     * t;
        float dy = (z0[2 * u + 1] - z0[2 * v + 1]) + (v0[2 * u + 1] - v0[2 * v + 1]) * t;
        acc += b - sqrtf(dx * dx + dy * dy);
    }
    red[threadIdx.x] = acc;
    __syncthreads();
    for (int s = 128; s > 0; s >>= 1) {
        if (threadIdx.x < s) red[threadIdx.x] += red[threadIdx.x + s];
        __syncthreads();
    }
    if (threadIdx.x == 0) partial[blockIdx.x] = red[0];
}

// ---------------------------------------------------------------------------
// Non-event term: one wave per 16x16 tile of the pair matrix.
// Three V_WMMA_F32_16X16X4_F32 compute the cross Grams; then per element:
// d^2(t) = q0 + t*q1 + t^2*q2  ->  2 FMA + sqrt + exp per Riemann sample.
// Tile partial = sum of exp(-d); global scale exp(b)*dt applied at the end.
// ---------------------------------------------------------------------------
__global__ void __launch_bounds__(32)
tile_kernel(const float* __restrict__ z0, const float* __restrict__ v0,
            const float* __restrict__ s0, const float* __restrict__ s1,
            const float* __restrict__ s2,
            const float* __restrict__ t0p, const float* __restrict__ tnp,
            float* __restrict__ partial) {
    int tj = blockIdx.x, ti = blockIdx.y;
    int bid = ti * TILES + tj;
    if (tj < ti) {                       // lower triangle: nothing to do
        if (threadIdx.x == 0) partial[bid] = 0.f;
        return;
    }
    int lane = threadIdx.x;
    int la = lane & 15;
    bool hi = lane >= 16;                // lanes 16-31 carry K2/K3 halves
    int irow = ti * 16 + la;
    int jcol = tj * 16 + la;

    float zix = z0[2 * irow], ziy = z0[2 * irow + 1];
    float vix = v0[2 * irow], viy = v0[2 * irow + 1];
    float zjx = z0[2 * jcol], zjy = z0[2 * jcol + 1];
    float vjx = v0[2 * jcol], vjy = v0[2 * jcol + 1];

    // A (16x4): lanes 0-15 -> K0,K1 ; lanes 16-31 -> K2,K3  (ISA 7.12.2)
    v2f Azz = { hi ? 0.f : zix, hi ? 0.f : ziy };            // z_i in K0,K1
    v2f Avv = { hi ? 0.f : vix, hi ? 0.f : viy };            // v_i in K0,K1
    v2f Ac  = { hi ? vix : zix, hi ? viy : ziy };            // z_i | v_i
    // B (4x16): lanes 0-15 -> rows K0,K1 ; lanes 16-31 -> rows K2,K3
    v2f Bzz = { hi ? 0.f : zjx, hi ? 0.f : zjy };            // z_j
    v2f Bvv = { hi ? 0.f : vjx, hi ? 0.f : vjy };            // v_j
    v2f Bc  = { hi ? zjx : vjx, hi ? zjy : vjy };            // v_j | z_j

    v8f cz = {};
    // Gzz[i][j] = z_i . z_j ; Gc = z_i.v_j + v_i.z_j ; Gvv = v_i . v_j
    v8f Gzz = __builtin_amdgcn_wmma_f32_16x16x4_f32(false, Azz, false, Bzz,
                                                    (short)0, cz, false, false);
    v8f Gc  = __builtin_amdgcn_wmma_f32_16x16x4_f32(false, Ac,  false, Bc,
                                                    (short)0, cz, false, false);
    v8f Gvv = __builtin_amdgcn_wmma_f32_16x16x4_f32(false, Avv, false, Bvv,
                                                    (short)0, cz, false, false);

    float t0s = t0p[0], tns = tnp[0];
    float dt = (tns - t0s) * (1.f / RSAMP);
    float sj0 = s0[jcol], sj1 = s1[jcol], sj2 = s2[jcol];

    float acc = 0.f;
#pragma unroll
    for (int r = 0; r < 8; ++r) {
        int ig = ti * 16 + (hi ? 8 : 0) + r;     // C/D: M = r + 8*(lane>=16)
        if (ig < jcol) {                         // strict upper triangle
            float q0 = s0[ig] + sj0 - 2.f * Gzz[r];
            float q1 = s1[ig] + sj1 - 2.f * Gc[r];
            float q2 = s2[ig] + sj2 - 2.f * Gvv[r];
#pragma unroll
            for (int k = 0; k < RSAMP; ++k) {
                float tm = t0s + (k + 0.5f) * dt;
                float d2 = fmaf(fmaf(q2, tm, q1), tm, q0);
                d2 = fmaxf(d2, 0.f);
                acc += __expf(-sqrtf(d2));
            }
        }
    }
    // wave32 reduction
    for (int off = 16; off > 0; off >>= 1) acc += __shfl_down(acc, off, 32);
    if (lane == 0) partial[bid] = acc;
}

// ---------------------------------------------------------------------------
// Final fixed-order reduction: out = eventSum - dt * exp(b) * tileSum
// ---------------------------------------------------------------------------
__global__ void final_kernel(const float* __restrict__ tileP,
                             const float* __restrict__ evP,
                             const float* __restrict__ beta,
                             const float* __restrict__ t0p,
                             const float* __restrict__ tnp,
                             float* __restrict__ out) {
    __shared__ float redA[256];
    __shared__ float redB[256];
    float a = 0.f, b = 0.f;
    for (int i = threadIdx.x; i < NTILEBLK; i += 256) a += tileP[i];
    for (int i = threadIdx.x; i < NEVBLK; i += 256) b += evP[i];
    redA[threadIdx.x] = a;
    redB[threadIdx.x] = b;
    __syncthreads();
    for (int s = 128; s > 0; s >>= 1) {
        if (threadIdx.x < s) {
            redA[threadIdx.x] += redA[threadIdx.x + s];
            redB[threadIdx.x] += redB[threadIdx.x + s];
        }
        __syncthreads();
    }
    if (threadIdx.x == 0) {
        float bb = beta[0];
        float dt = (tnp[0] - t0p[0]) * (1.f / RSAMP);
        out[0] = redB[0] - dt * __expf(bb) * redA[0];
    }
}

// ---------------------------------------------------------------------------
extern "C" void kernel_launch(void* const* d_in, const int* in_sizes, int n_in,
                              void* d_out, int out_size, void* d_ws, size_t ws_size,
                              hipStream_t stream) {
    const float* beta   = (const float*)d_in[0];
    const float* z0     = (const float*)d_in[1];
    const float* v0     = (const float*)d_in[2];
    const int*   uv     = (const int*)d_in[3];
    const float* data_t = (const float*)d_in[4];
    const float* t0     = (const float*)d_in[5];
    const float* tn     = (const float*)d_in[6];
    float* out = (float*)d_out;

    float* ws    = (float*)d_ws;
    float* tileP = ws;                       // NTILEBLK floats
    float* evP   = tileP + NTILEBLK;         // NEVBLK floats
    float* s0    = evP + NEVBLK;             // N_POINTS floats
    float* s1    = s0 + N_POINTS;
    float* s2    = s1 + N_POINTS;

    stats_kernel<<<(N_POINTS + 255) / 256, 256, 0, stream>>>(z0, v0, s0, s1, s2);
    event_kernel<<<NEVBLK, 256, 0, stream>>>(beta, z0, v0, uv, data_t, evP);
    tile_kernel<<<dim3(TILES, TILES), 32, 0, stream>>>(z0, v0, s0, s1, s2,
                                                       t0, tn, tileP);
    final_kernel<<<1, 256, 0, stream>>>(tileP, evP, beta, t0, tn, out);
    (void)in_sizes; (void)n_in; (void)out_size; (void)ws_size;
}